// Schema_Relation_Network_64415919506090
// MI455X (gfx1250) — compile-verified
//
#include <hip/hip_runtime.h>
#include <math.h>

#define D 128
#define RELS 3

typedef float v2f __attribute__((ext_vector_type(2)));
typedef float v8f __attribute__((ext_vector_type(8)));

// ---------------- small helpers ----------------

__device__ __forceinline__ void atomicMaxF(float* addr, float value) {
  // works for all finite floats + -inf init: positives via signed-int max,
  // negatives via unsigned-int min (IEEE754 bit ordering trick)
  if (value >= 0.f) atomicMax((int*)addr, __float_as_int(value));
  else              atomicMin((unsigned int*)addr, __float_as_uint(value));
}

__device__ __forceinline__ float lrelu(float x) { return x > 0.f ? x : 0.2f * x; }

// ---------------- init kernels ----------------

__global__ void k_init_wsum(float* wsum) {
  if (threadIdx.x < RELS) wsum[threadIdx.x] = 0.f;
}

__global__ void k_init_rel(float* __restrict__ zacc, float* __restrict__ m,
                           float* __restrict__ zsum, int nfeat, int nnode) {
  int i = blockIdx.x * blockDim.x + threadIdx.x;
  if (i < nfeat) zacc[i] = 0.f;
  if (i < nnode) { m[i] = -__builtin_inff(); zsum[i] = 0.f; }
}

// wa[d] = sum_h W[d][h]*al[h] ; wr[d] = sum_h W[d][h]*ar[h]
__global__ __launch_bounds__(256) void k_rowdot(const float* __restrict__ W,
                                                const float* __restrict__ al,
                                                const float* __restrict__ ar,
                                                float* __restrict__ wa,
                                                float* __restrict__ wr) {
  int t = threadIdx.x;
  const float* v = (t < D) ? al : ar;
  float* o = (t < D) ? wa : wr;
  int d = t & (D - 1);
  float s = 0.f;
#pragma unroll 8
  for (int h = 0; h < D; ++h) s += W[d * D + h] * v[h];
  o[d] = s;
}

// ---------------- WMMA fp32 GEMM: H[N x 128] = X[N x 128] @ W[128 x 128] ----------------
// block = 256 threads = 8 waves; each wave computes a 16-row stripe.
// A fragments (16x4 fp32 tile): lane l holds A[l%16][k0 + {0,1}], k0 = kt*4 + 2*(l/16)
// B fragments (4x16 fp32 tile): lane l holds B[k0 + {0,1}][l%16]
// C/D (16x16 fp32): acc[v] = D[v + 8*(l/16)][l%16]
__global__ __launch_bounds__(256) void k_gemm_wmma(const float* __restrict__ X,
                                                   const float* __restrict__ W,
                                                   float* __restrict__ H, int N) {
  __shared__ float Ws[D * D];  // 64 KB, row-major [k][n]
  int t = threadIdx.x;
#pragma unroll
  for (int i = 0; i < (D * D / 4) / 256; ++i)
    ((float4*)Ws)[t + i * 256] = ((const float4*)W)[t + i * 256];
  __syncthreads();

  const int wave = t >> 5;
  const int lane = t & 31;
  const int lrow = lane & 15;
  const int half = lane >> 4;
  const int rowbase = blockIdx.x * 128 + wave * 16;

  int xr = rowbase + lrow;
  if (xr > N - 1) xr = N - 1;  // clamp; masked at store
  const float* xp = X + (size_t)xr * D;

  v2f afrag[32];
#pragma unroll
  for (int kt = 0; kt < 32; ++kt) {
    int k0 = kt * 4 + 2 * half;
    afrag[kt] = *(const v2f*)(xp + k0);  // 8B aligned (k0 even)
  }

#pragma unroll
  for (int nt = 0; nt < 8; ++nt) {
    v8f acc = {};
    int nb = nt * 16 + lrow;
#pragma unroll
    for (int kt = 0; kt < 32; ++kt) {
      int k0 = kt * 4 + 2 * half;
      v2f b;
      b.x = Ws[k0 * D + nb];
      b.y = Ws[(k0 + 1) * D + nb];
      acc = __builtin_amdgcn_wmma_f32_16x16x4_f32(false, afrag[kt], false, b,
                                                  (short)0, acc, false, false);
    }
#pragma unroll
    for (int v = 0; v < 8; ++v) {
      int grow = rowbase + v + 8 * half;
      if (grow < N) H[(size_t)grow * D + nb] = acc[v];
    }
  }
}

// ---------------- matvec: out[i] = X[i,:] . v  (one wave per row) ----------------
__global__ __launch_bounds__(256) void k_matvec(const float* __restrict__ X,
                                                const float* __restrict__ v,
                                                float* __restrict__ out, int n) {
  int wave = threadIdx.x >> 5, lane = threadIdx.x & 31;
  int row = blockIdx.x * 8 + wave;
  if (row >= n) return;
  float4 x = ((const float4*)X)[(size_t)row * (D / 4) + lane];
  float4 c = ((const float4*)v)[lane];
  float s = x.x * c.x + x.y * c.y + x.z * c.z + x.w * c.w;
#pragma unroll
  for (int off = 16; off; off >>= 1) s += __shfl_xor(s, off, 32);
  if (lane == 0) out[row] = s;
}

// ---------------- edge softmax passes ----------------

__global__ void k_edge_max(const int* __restrict__ si, const int* __restrict__ di,
                           const float* __restrict__ el, const float* __restrict__ er,
                           float* __restrict__ m, int E) {
  int i = blockIdx.x * blockDim.x + threadIdx.x;
  if (i >= E) return;
  int d = di[i];
  atomicMaxF(&m[d], lrelu(el[si[i]] + er[d]));
}

__global__ void k_edge_sum(const int* __restrict__ si, const int* __restrict__ di,
                           const float* __restrict__ el, const float* __restrict__ er,
                           const float* __restrict__ m, float* __restrict__ zsum, int E) {
  int i = blockIdx.x * blockDim.x + threadIdx.x;
  if (i >= E) return;
  int d = di[i];
  float e = lrelu(el[si[i]] + er[d]);
  atomicAdd(&zsum[d], expf(e - m[d]));
}

// one wave per edge: zacc[d,:] += alpha * hs[s,:]
__global__ __launch_bounds__(256) void k_edge_aggr(const int* __restrict__ si,
                                                   const int* __restrict__ di,
                                                   const float* __restrict__ el,
                                                   const float* __restrict__ er,
                                                   const float* __restrict__ m,
                                                   const float* __restrict__ zsum,
                                                   const float* __restrict__ hs,
                                                   float* __restrict__ zacc, int E) {
  int wave = threadIdx.x >> 5, lane = threadIdx.x & 31;
  int e = blockIdx.x * 8 + wave;
  if (e >= E) return;
  int s = si[e], d = di[e];
  float x = lrelu(el[s] + er[d]);
  float alpha = expf(x - m[d]) / zsum[d];
  float4 h = ((const float4*)hs)[(size_t)s * (D / 4) + lane];
  float* zp = zacc + (size_t)d * D + lane * 4;
  atomicAdd(zp + 0, alpha * h.x);
  atomicAdd(zp + 1, alpha * h.y);
  atomicAdd(zp + 2, alpha * h.z);
  atomicAdd(zp + 3, alpha * h.w);
}

// z = elu(zacc + b), in place
__global__ void k_fin(float* __restrict__ Z, const float* __restrict__ b, int n) {
  int i = blockIdx.x * blockDim.x + threadIdx.x;
  if (i >= n) return;
  float x = Z[i] + b[i & (D - 1)];
  Z[i] = x > 0.f ? x : expf(x) - 1.f;
}

// semantic attention pre-softmax: wsum += sum_i tanh(Z[i,:]@W1 + b1) @ W2
__global__ __launch_bounds__(128) void k_sem(const float* __restrict__ Z,
                                             const float* __restrict__ W1,
                                             const float* __restrict__ b1,
                                             const float* __restrict__ W2,
                                             float* __restrict__ wsum) {
  __shared__ float zs[D];
  __shared__ float red[D];
  int node = blockIdx.x;
  int h = threadIdx.x;
  zs[h] = Z[(size_t)node * D + h];
  __syncthreads();
  float acc = b1[h];
#pragma unroll 8
  for (int d = 0; d < D; ++d) acc += zs[d] * W1[d * D + h];
  red[h] = tanhf(acc) * W2[h];
  __syncthreads();
  for (int s = 64; s; s >>= 1) {
    if (h < s) red[h] += red[h + s];
    __syncthreads();
  }
  if (h == 0) atomicAdd(wsum, red[0]);
}

// final mix: softmax over mean-scores, weighted sum, write z + att_mp tail
__global__ void k_mix(const float* __restrict__ Z3, const float* __restrict__ wsum,
                      float* __restrict__ out, int n, float invN) {
  int i = blockIdx.x * blockDim.x + threadIdx.x;
  float w0 = wsum[0] * invN, w1 = wsum[1] * invN, w2 = wsum[2] * invN;
  float mx = fmaxf(w0, fmaxf(w1, w2));
  float e0 = expf(w0 - mx), e1 = expf(w1 - mx), e2 = expf(w2 - mx);
  float inv = 1.f / (e0 + e1 + e2);
  float a0 = e0 * inv, a1 = e1 * inv, a2 = e2 * inv;
  if (i < n)
    out[i] = a0 * Z3[i] + a1 * Z3[(size_t)n + i] + a2 * Z3[2 * (size_t)n + i];
  if (i == 0) { out[n] = a0; out[n + 1] = a1; out[n + 2] = a2; }
}

// ---------------- launch ----------------

extern "C" void kernel_launch(void* const* d_in, const int* in_sizes, int n_in,
                              void* d_out, int out_size, void* d_ws, size_t ws_size,
                              hipStream_t stream) {
  const float* dst_feat = (const float*)d_in[0];
  const float* src_feat[RELS] = {(const float*)d_in[1], (const float*)d_in[2],
                                 (const float*)d_in[3]};
  const float* sem_W1 = (const float*)d_in[4];
  const float* sem_b1 = (const float*)d_in[5];
  const float* sem_W2 = (const float*)d_in[6];
  const float *Wg[RELS], *al[RELS], *ar[RELS], *bg[RELS];
  const int *sidx[RELS], *didx[RELS];
  for (int r = 0; r < RELS; ++r) {
    int b = 7 + 6 * r;
    Wg[r] = (const float*)d_in[b + 0];
    al[r] = (const float*)d_in[b + 1];
    ar[r] = (const float*)d_in[b + 2];
    bg[r] = (const float*)d_in[b + 3];
    sidx[r] = (const int*)d_in[b + 4];
    didx[r] = (const int*)d_in[b + 5];
  }
  const int ND = in_sizes[0] / D;  // 50000
  const int NS = in_sizes[1] / D;  // 50000
  const int E  = in_sizes[11];     // 800000

  // workspace layout (floats)
  float* ws   = (float*)d_ws;
  float* z3   = ws;                              // RELS * ND * D
  float* hs   = z3 + (size_t)RELS * ND * D;      // NS * D (reused per relation)
  float* el   = hs + (size_t)NS * D;             // NS
  float* er   = el + NS;                         // ND
  float* m    = er + ND;                         // ND
  float* zsum = m + ND;                          // ND
  float* wa   = zsum + ND;                       // D
  float* wr   = wa + D;                          // D
  float* wsum = wr + D;                          // RELS

  k_init_wsum<<<1, 32, 0, stream>>>(wsum);

  for (int r = 0; r < RELS; ++r) {
    float* zr = z3 + (size_t)r * ND * D;
    k_rowdot<<<1, 256, 0, stream>>>(Wg[r], al[r], ar[r], wa, wr);
    k_init_rel<<<(ND * D + 255) / 256, 256, 0, stream>>>(zr, m, zsum, ND * D, ND);
    k_gemm_wmma<<<(NS + 127) / 128, 256, 0, stream>>>(src_feat[r], Wg[r], hs, NS);
    k_matvec<<<(NS + 7) / 8, 256, 0, stream>>>(src_feat[r], wa, el, NS);
    k_matvec<<<(ND + 7) / 8, 256, 0, stream>>>(dst_feat, wr, er, ND);
    k_edge_max<<<(E + 255) / 256, 256, 0, stream>>>(sidx[r], didx[r], el, er, m, E);
    k_edge_sum<<<(E + 255) / 256, 256, 0, stream>>>(sidx[r], didx[r], el, er, m, zsum, E);
    k_edge_aggr<<<(E + 7) / 8, 256, 0, stream>>>(sidx[r], didx[r], el, er, m, zsum, hs,
                                                 zr, E);
    k_fin<<<(ND * D + 255) / 256, 256, 0, stream>>>(zr, bg[r], ND * D);
    k_sem<<<ND, 128, 0, stream>>>(zr, sem_W1, sem_b1, sem_W2, wsum + r);
  }

  k_mix<<<(ND * D + 255) / 256, 256, 0, stream>>>(z3, wsum, (float*)d_out, ND * D,
                                                  1.0f / (float)ND);
}